// DISTANA_62886911148706
// MI455X (gfx1250) — compile-verified
//
#include <hip/hip_runtime.h>

// fp32 WMMA fragments: 16x16x4 f32 -> A,B are 2 VGPRs (v2f), C/D are 8 VGPRs (v8f)
typedef __attribute__((ext_vector_type(2))) float v2f;
typedef __attribute__((ext_vector_type(8))) float v8f;

#define ROWS_PER_BLOCK 128
#define NTHREADS 256
#define IN_DIM 9
#define IN_PAD 12   // pad K to multiple of 4 (zeros)
#define PRE 16
#define CELLS 32
#define OUT_DIM 9
#define OUT_PAD 16  // pad N to 16 (zeros)

// Branch-free sigmoid/tanh entirely on the hardware trans pipe:
// v_exp_f32 + v_rcp_f32 (raw hw reciprocal, ~1ulp -- matches the exp accuracy floor).
// tanh(x) = 1 - 2/(exp(2x)+1): exp->0 gives -1, exp->inf gives +1, no branches.
__device__ __forceinline__ float fast_sigmoid(float x) {
    return __builtin_amdgcn_rcpf(1.0f + __expf(-x));
}
__device__ __forceinline__ float fast_tanh(float x) {
    float e = __expf(2.0f * x);
    return __builtin_fmaf(-2.0f, __builtin_amdgcn_rcpf(e + 1.0f), 1.0f);
}

__global__ __launch_bounds__(NTHREADS) void DISTANA_62886911148706_kernel(
    const float* __restrict__ input,     // [B, 9]
    const float* __restrict__ old_h,     // [B, 32]
    const float* __restrict__ old_cell,  // [B, 32]
    const float* __restrict__ pre_w,     // [9, 16]
    const float* __restrict__ lstm_w,    // [16, 32]
    const float* __restrict__ post_w,    // [32, 9]
    float* __restrict__ out,             // [B, 9]
    float* __restrict__ new_h,           // [B, 32]
    float* __restrict__ new_cell,        // [B, 32]
    int nrows)
{
    __shared__ float sWpre[IN_PAD * PRE];          // 12x16, rows 9..11 zero
    __shared__ float sWlstm[PRE * CELLS];          // 16x32
    __shared__ float sWpost[CELLS * OUT_PAD];      // 32x16, cols 9..15 zero
    __shared__ float sIn[ROWS_PER_BLOCK * IN_PAD]; // 128x12 input tile (zero-padded K)
    __shared__ float sAct[ROWS_PER_BLOCK * PRE];   // 128x16 pre-activation (A-layout restage)
    __shared__ float sH[ROWS_PER_BLOCK * CELLS];   // 128x32 new_h (A-layout restage)

    const int tid  = threadIdx.x;
    const int row0 = blockIdx.x * ROWS_PER_BLOCK;

    // ---- cooperative staging (coalesced) ----
    for (int i = tid; i < IN_PAD * PRE; i += NTHREADS) {
        int k = i / PRE, n = i % PRE;
        sWpre[i] = (k < IN_DIM) ? pre_w[k * PRE + n] : 0.0f;
    }
    for (int i = tid; i < PRE * CELLS; i += NTHREADS)
        sWlstm[i] = lstm_w[i];
    for (int i = tid; i < CELLS * OUT_PAD; i += NTHREADS) {
        int k = i >> 4, n = i & 15;
        sWpost[i] = (n < OUT_DIM) ? post_w[k * OUT_DIM + n] : 0.0f;
    }
    for (int i = tid; i < ROWS_PER_BLOCK * IN_PAD; i += NTHREADS) {
        int r = i / IN_PAD, c = i % IN_PAD;
        int gr = row0 + r;
        sIn[i] = (c < IN_DIM && gr < nrows) ? input[(size_t)gr * IN_DIM + c] : 0.0f;
    }
    __syncthreads();

    // per-wave 16-row tile
    const int lane = tid & 31;
    const int ln16 = lane & 15;
    const int hi   = lane >> 4;        // 0: lanes 0-15, 1: lanes 16-31
    const int wrow = (tid >> 5) * 16;  // wave's local row base

    const v8f vzero = {0.f, 0.f, 0.f, 0.f, 0.f, 0.f, 0.f, 0.f};

    // ---- matmul 1: pre_act = tanh( in(16x12) x Wpre(12x16) ), 3 K-steps ----
    v8f acc = vzero;
#pragma unroll
    for (int s = 0; s < 3; ++s) {
        const int kb = 4 * s + 2 * hi;   // per-lane K base (A and B layouts mirror)
        v2f a = *(const v2f*)&sIn[(wrow + ln16) * IN_PAD + kb];
        v2f b;
        b.x = sWpre[(kb + 0) * PRE + ln16];
        b.y = sWpre[(kb + 1) * PRE + ln16];
        acc = __builtin_amdgcn_wmma_f32_16x16x4_f32(false, a, false, b, (short)0, acc,
                                                    false, false);
    }
    // C/D element r of lane = (row r+8*hi, col ln16); restage as A-layout in LDS
#pragma unroll
    for (int r = 0; r < 8; ++r)
        sAct[(wrow + r + 8 * hi) * PRE + ln16] = fast_tanh(acc[r]);
    __syncthreads();

    // ---- matmul 2: lstm_in(16x32) = pre_act(16x16) x Wlstm(16x32), 2 N-tiles x 4 K-steps ----
    v8f c0 = vzero, c1 = vzero;
#pragma unroll
    for (int s = 0; s < 4; ++s) {
        const int kb = 4 * s + 2 * hi;
        v2f a = *(const v2f*)&sAct[(wrow + ln16) * PRE + kb];
        v2f b0, b1;
        b0.x = sWlstm[(kb + 0) * CELLS + ln16];
        b0.y = sWlstm[(kb + 1) * CELLS + ln16];
        b1.x = sWlstm[(kb + 0) * CELLS + 16 + ln16];
        b1.y = sWlstm[(kb + 1) * CELLS + 16 + ln16];
        c0 = __builtin_amdgcn_wmma_f32_16x16x4_f32(false, a, false, b0, (short)0, c0,
                                                   false, false);
        c1 = __builtin_amdgcn_wmma_f32_16x16x4_f32(false, a, false, b1, (short)0, c1,
                                                   false, false);
    }

    // ---- gating (elementwise in C/D layout; global access is 2x64B contiguous per step) ----
#pragma unroll
    for (int r = 0; r < 8; ++r) {
        const int lrow = wrow + r + 8 * hi;
        const int grow = row0 + lrow;
        if (grow < nrows) {
            const size_t gb = (size_t)grow * CELLS;
            {   // cells [0,16)
                float co = old_cell[gb + ln16];
                float ho = old_h[gb + ln16];
                float x  = c0[r];
                float f  = fast_sigmoid(x + co);
                float g  = fast_tanh(x + ho);
                float nc = f * co + (1.0f - f) * g;
                float nh = fast_tanh(nc);
                new_cell[gb + ln16] = nc;
                new_h[gb + ln16]    = nh;
                sH[lrow * CELLS + ln16] = nh;
            }
            {   // cells [16,32)
                float co = old_cell[gb + 16 + ln16];
                float ho = old_h[gb + 16 + ln16];
                float x  = c1[r];
                float f  = fast_sigmoid(x + co);
                float g  = fast_tanh(x + ho);
                float nc = f * co + (1.0f - f) * g;
                float nh = fast_tanh(nc);
                new_cell[gb + 16 + ln16] = nc;
                new_h[gb + 16 + ln16]    = nh;
                sH[lrow * CELLS + 16 + ln16] = nh;
            }
        } else {
            sH[lrow * CELLS + ln16] = 0.0f;
            sH[lrow * CELLS + 16 + ln16] = 0.0f;
        }
    }
    __syncthreads();

    // ---- matmul 3: out(16x16pad) = new_h(16x32) x Wpost(32x16pad), 8 K-steps ----
    v8f c2 = vzero;
#pragma unroll
    for (int s = 0; s < 8; ++s) {
        const int kb = 4 * s + 2 * hi;
        v2f a = *(const v2f*)&sH[(wrow + ln16) * CELLS + kb];
        v2f b;
        b.x = sWpost[(kb + 0) * OUT_PAD + ln16];
        b.y = sWpost[(kb + 1) * OUT_PAD + ln16];
        c2 = __builtin_amdgcn_wmma_f32_16x16x4_f32(false, a, false, b, (short)0, c2,
                                                   false, false);
    }
    if (ln16 < OUT_DIM) {
#pragma unroll
        for (int r = 0; r < 8; ++r) {
            const int grow = row0 + wrow + r + 8 * hi;
            if (grow < nrows)
                out[(size_t)grow * OUT_DIM + ln16] = c2[r];
        }
    }
}

extern "C" void kernel_launch(void* const* d_in, const int* in_sizes, int n_in,
                              void* d_out, int out_size, void* d_ws, size_t ws_size,
                              hipStream_t stream) {
    const float* input    = (const float*)d_in[0];
    const float* old_h    = (const float*)d_in[1];
    const float* old_cell = (const float*)d_in[2];
    const float* pre_w    = (const float*)d_in[3];
    const float* lstm_w   = (const float*)d_in[4];
    const float* post_w   = (const float*)d_in[5];

    const int nrows = in_sizes[0] / IN_DIM;  // B

    // d_out = concat(out[B,9], new_h[B,32], new_cell[B,32])
    float* out_p = (float*)d_out;
    float* nh_p  = out_p + (size_t)nrows * OUT_DIM;
    float* nc_p  = nh_p + (size_t)nrows * CELLS;

    const int nblocks = (nrows + ROWS_PER_BLOCK - 1) / ROWS_PER_BLOCK;
    DISTANA_62886911148706_kernel<<<nblocks, NTHREADS, 0, stream>>>(
        input, old_h, old_cell, pre_w, lstm_w, post_w, out_p, nh_p, nc_p, nrows);
}